// LambdaRankLoss_9637906612627
// MI455X (gfx1250) — compile-verified
//
#include <hip/hip_runtime.h>
#include <hip/hip_bf16.h>
#include <stdint.h>

typedef __attribute__((ext_vector_type(2))) float v2f;
typedef __attribute__((ext_vector_type(8))) float v8f;

#define SIGMA 1.0f
#define LOG2E 1.44269504088896340736f
#define LN2   0.69314718055994530942f

// ---------------------------------------------------------------------------
// Kernel 1: per-batch-row prep. One 1024-thread block per row.
//  - rank[i]  = 1 + #{j: s_j > s_i} + #{j: s_j == s_i && j < i}   (stable argsort)
//  - irank[i] = same counting on labels (monotone with gains)
//  - g[i] = 2^y - 1 ; d[i] = 1/log2(rank[i]+1) ; inv_idcg = 1/max(IDCG,1e-8)
// LDS holds (s,y) packed as float2 -> one ds_load_b64 per compare.
// ---------------------------------------------------------------------------
__global__ __launch_bounds__(1024) void lr_prep(const float* __restrict__ logits,
                                                const float* __restrict__ labels,
                                                float* __restrict__ gbuf,
                                                float* __restrict__ dbuf,
                                                float* __restrict__ inv_idcg,
                                                int Nn) {
  __shared__ float2 sy[1024];
  __shared__ float red[32];
  const int b = blockIdx.x;
  const int i = threadIdx.x;
  const float si = logits[b * Nn + i];
  const float yi = labels[b * Nn + i];
  sy[i] = make_float2(si, yi);
  __syncthreads();

  int rank = 0, irank = 0;
  #pragma unroll 8
  for (int j = 0; j < Nn; ++j) {
    const float2 v = sy[j];
    const bool jlt = (j < i);
    rank  += ((v.x > si) || ((v.x == si) && jlt)) ? 1 : 0;
    irank += ((v.y > yi) || ((v.y == yi) && jlt)) ? 1 : 0;
  }

  const float gi = exp2f(yi) - 1.0f;
  // rank args are in [2, N+1]: safe for the raw v_log_f32
  const float di = 1.0f / __builtin_amdgcn_logf((float)(rank + 2));
  gbuf[b * Nn + i] = gi;
  dbuf[b * Nn + i] = di;

  // IDCG contribution: gains at ideal (descending-gain) position
  float v = gi / __builtin_amdgcn_logf((float)(irank + 2));
  #pragma unroll
  for (int off = 16; off > 0; off >>= 1) v += __shfl_down(v, off, 32);
  if ((threadIdx.x & 31) == 0) red[threadIdx.x >> 5] = v;
  __syncthreads();
  if (threadIdx.x < 32) {
    float t = red[threadIdx.x];
    #pragma unroll
    for (int off = 16; off > 0; off >>= 1) t += __shfl_down(t, off, 32);
    if (threadIdx.x == 0) inv_idcg[b] = 1.0f / fmaxf(t, 1e-8f);
  }
}

// ---------------------------------------------------------------------------
// Kernel 2: pair loss over full NxN (symmetric; caller halves at the end).
// Block = (b, i-tile of 16 rows), 8 waves; each wave owns 8 j-tiles.
// Outer differences via V_WMMA_F32_16X16X4_F32:
//   A row = [v_i, 1, 0, 0]  (K=0..3),  B col = [1, -v_j, 0, 0]  ->  D = v_i - v_j
// Epilogue (sign-free, raw-transcendental):
//   softplus(-sign(gd)*sd)*|gd| = max(-sd*gd,0) + ln2*log2(1+2^(-log2e*|sd|))*|gd|
// Log argument is in (1,2] -> bare v_log_f32, no denormal fixup needed.
// ---------------------------------------------------------------------------
__global__ __launch_bounds__(256) void lr_pairs(const float* __restrict__ logits,
                                                const float* __restrict__ gbuf,
                                                const float* __restrict__ dbuf,
                                                const unsigned char* __restrict__ mask,
                                                const float* __restrict__ inv_idcg,
                                                float* __restrict__ partial,
                                                int Nn) {
  const int b    = blockIdx.y;
  const int i0   = blockIdx.x * 16;
  const int lane = threadIdx.x & 31;
  const int w    = threadIdx.x >> 5;
  const int col  = lane & 15;
  const bool lo  = lane < 16;

  const float* __restrict__ srow = logits + (size_t)b * Nn;
  const float* __restrict__ grow = gbuf   + (size_t)b * Nn;
  const float* __restrict__ drow = dbuf   + (size_t)b * Nn;
  const unsigned char* __restrict__ mrow = mask + (size_t)b * Nn;

  // A operands for this i-tile (shared by all 8 j-tiles of this wave)
  const float oneA = lo ? 1.0f : 0.0f;
  const v2f As = { lo ? srow[i0 + col] : 0.0f, oneA };
  const v2f Ag = { lo ? grow[i0 + col] : 0.0f, oneA };
  const v2f Ad = { lo ? drow[i0 + col] : 0.0f, oneA };

  // Row masks, hoisted to 8 block-invariant floats (no in-loop cvt/shift).
  const uint32_t* mrow32 = (const uint32_t*)(mrow + i0);
  const uint32_t mi01 = mrow32[0], mi11 = mrow32[1];
  const uint32_t mi21 = mrow32[2], mi31 = mrow32[3];
  float mirow[8];
  #pragma unroll
  for (int r = 0; r < 8; ++r) {
    const uint32_t mw = lo ? ((r < 4) ? mi01 : mi11) : ((r < 4) ? mi21 : mi31);
    mirow[r] = (float)((mw >> ((r & 3) * 8)) & 0xFFu);
  }

  // Hoist all B operands + column masks for this wave's 8 j-tiles.
  v2f Bsv[8], Bgv[8], Bdv[8];
  float mjv[8];
  #pragma unroll
  for (int t = 0; t < 8; ++t) {
    const int j0 = (w + 8 * t) * 16;
    Bsv[t] = (v2f){ oneA, lo ? -srow[j0 + col] : 0.0f };
    Bgv[t] = (v2f){ oneA, lo ? -grow[j0 + col] : 0.0f };
    Bdv[t] = (v2f){ oneA, lo ? -drow[j0 + col] : 0.0f };
    mjv[t] = (float)mrow[j0 + col];
  }

  float acc = 0.0f;
  #pragma unroll
  for (int t = 0; t < 8; ++t) {
    const v8f zero = {};
    // D = v_i - v_j in the 16x16 accumulator layout
    v8f SD = __builtin_amdgcn_wmma_f32_16x16x4_f32(false, As, false, Bsv[t],
                                                   (short)0, zero, false, false);
    v8f GD = __builtin_amdgcn_wmma_f32_16x16x4_f32(false, Ag, false, Bgv[t],
                                                   (short)0, zero, false, false);
    v8f DD = __builtin_amdgcn_wmma_f32_16x16x4_f32(false, Ad, false, Bdv[t],
                                                   (short)0, zero, false, false);
    const float mj = mjv[t];
    #pragma unroll
    for (int r = 0; r < 8; ++r) {
      const float sd = SIGMA * SD[r];
      const float gd = GD[r];
      const float dd = DD[r];
      // softplus(-sign(gd)*sd)*|gd| = max(-sd*gd,0) + ln2*log2(1+e)*|gd|
      const float lin   = fmaxf(-sd * gd, 0.0f);
      const float e     = __builtin_amdgcn_exp2f(-LOG2E * fabsf(sd)); // (0,1]
      const float log2v = __builtin_amdgcn_logf(1.0f + e);            // arg in (1,2]
      const float term  = fmaf(log2v, LN2 * fabsf(gd), lin);
      acc = fmaf(term, fabsf(dd) * (mirow[r] * mj), acc);
    }
  }
  acc *= inv_idcg[b];   // constant over the block: one mul instead of 33M

  // Deterministic block reduction -> one partial per block
  #pragma unroll
  for (int off = 16; off > 0; off >>= 1) acc += __shfl_down(acc, off, 32);
  __shared__ float red[8];
  if (lane == 0) red[w] = acc;
  __syncthreads();
  if (threadIdx.x == 0) {
    float tsum = 0.0f;
    #pragma unroll
    for (int k = 0; k < 8; ++k) tsum += red[k];
    partial[(size_t)blockIdx.y * gridDim.x + blockIdx.x] = tsum;
  }
}

// ---------------------------------------------------------------------------
// Kernel 3: deterministic final reduction; scale = 0.5 (i<j symmetry) / B (mean)
// ---------------------------------------------------------------------------
__global__ __launch_bounds__(256) void lr_finalize(const float* __restrict__ partial,
                                                   float* __restrict__ out,
                                                   int n, float scale) {
  __shared__ float red[8];
  const int tid = threadIdx.x;
  float v = 0.0f;
  for (int k = tid; k < n; k += 256) v += partial[k];
  #pragma unroll
  for (int off = 16; off > 0; off >>= 1) v += __shfl_down(v, off, 32);
  if ((tid & 31) == 0) red[tid >> 5] = v;
  __syncthreads();
  if (tid == 0) {
    float t = 0.0f;
    #pragma unroll
    for (int k = 0; k < 8; ++k) t += red[k];
    out[0] = t * scale;
  }
}

extern "C" void kernel_launch(void* const* d_in, const int* in_sizes, int n_in,
                              void* d_out, int out_size, void* d_ws, size_t ws_size,
                              hipStream_t stream) {
  const float* logits = (const float*)d_in[0];               // (B, N, 1) f32
  const float* labels = (const float*)d_in[1];               // (B, N, 1) f32
  const unsigned char* mask = (const unsigned char*)d_in[2]; // (B, N) bool

  const int Nn = 1024;
  const int Bn = in_sizes[0] / Nn;

  float* gbuf     = (float*)d_ws;            // B*N
  float* dbuf     = gbuf + (size_t)Bn * Nn;  // B*N
  float* inv_idcg = dbuf + (size_t)Bn * Nn;  // B
  float* partial  = inv_idcg + Bn;           // B*(N/16)

  lr_prep<<<Bn, 1024, 0, stream>>>(logits, labels, gbuf, dbuf, inv_idcg, Nn);

  dim3 grid(Nn / 16, Bn);
  lr_pairs<<<grid, 256, 0, stream>>>(logits, gbuf, dbuf, mask, inv_idcg, partial, Nn);

  const int nPartial = Bn * (Nn / 16);
  lr_finalize<<<1, 256, 0, stream>>>(partial, (float*)d_out, nPartial,
                                     0.5f / (float)Bn);
}